// PETSystemModel_40321152975183
// MI455X (gfx1250) — compile-verified
//
#include <hip/hip_runtime.h>

// Radon transform (skimage-style) for 512x512 image, 180 integer-degree angles.
// Gather-bound: image (1 MB) fully L2-resident on MI455X (192 MB L2).
// Branch-free bilinear gathers (power-of-two wrap + cndmask zeroing),
// per-lane register accumulation + one V_WMMA_F32_16X16X4_F32 wave reduction.

typedef float v2f __attribute__((ext_vector_type(2)));
typedef float v8f __attribute__((ext_vector_type(8)));

#define N_IMG 512
#define N_ANG 180

__device__ __forceinline__ float bilin(const float* __restrict__ img, float sy, float sx) {
    float y0f = floorf(sy), x0f = floorf(sx);
    float wy = sy - y0f, wx = sx - x0f;
    int yi = (int)y0f, xi = (int)x0f;

    // Validity: single unsigned compare per coordinate.
    bool y0in = (unsigned)yi       < (unsigned)N_IMG;
    bool y1in = (unsigned)(yi + 1) < (unsigned)N_IMG;
    bool x0in = (unsigned)xi       < (unsigned)N_IMG;
    bool x1in = (unsigned)(xi + 1) < (unsigned)N_IMG;

    // Always-in-bounds wrapped addresses (N_IMG is a power of two); the
    // actual value is zeroed by the selects below when out of range, so any
    // in-bounds address is fine. This keeps the loads unconditional ->
    // straight-line code (v_cndmask), no EXEC-mask branches.
    int ry0 = (yi & (N_IMG - 1)) << 9;        // row offset, elements
    int ry1 = ((yi + 1) & (N_IMG - 1)) << 9;
    int cx0 = xi & (N_IMG - 1);
    int cx1 = (xi + 1) & (N_IMG - 1);

    float v00 = img[ry0 + cx0];
    float v01 = img[ry0 + cx1];
    float v10 = img[ry1 + cx0];
    float v11 = img[ry1 + cx1];

    v00 = (y0in && x0in) ? v00 : 0.0f;
    v01 = (y0in && x1in) ? v01 : 0.0f;
    v10 = (y1in && x0in) ? v10 : 0.0f;
    v11 = (y1in && x1in) ? v11 : 0.0f;

    float a = fmaf(wx, v01 - v00, v00);
    float b = fmaf(wx, v11 - v10, v10);
    return fmaf(wy, b - a, a);
}

__global__ void __launch_bounds__(256)
radon_wmma_kernel(const float* __restrict__ img, const int* __restrict__ angles,
                  float* __restrict__ out) {
    const int blocksPerAngle = N_IMG / 128;           // 4
    const int ang  = blockIdx.x / blocksPerAngle;
    const int dblk = blockIdx.x % blocksPerAngle;
    const int wave = threadIdx.x >> 5;
    const int lane = threadIdx.x & 31;
    const int det  = dblk * 128 + wave * 16 + (lane & 15);
    const int khalf = lane >> 4;                      // 0: K slots {0,1}, 1: {2,3}

    const float theta = (float)angles[ang] * 0.017453292519943295f;
    const float co = cosf(theta);
    const float si = sinf(theta);
    const float c  = (float)(N_IMG - 1) * 0.5f;       // 255.5

    // Fixed detector column x: sx = co*xc + si*yc + c ; sy = -si*xc + co*yc + c
    const float xc  = (float)det - c;
    const float sx0 = fmaf(co, xc, c);
    const float sy0 = fmaf(-si, xc, c);

    float ps0 = 0.0f;   // y = 4t + 2*khalf
    float ps1 = 0.0f;   // y = 4t + 2*khalf + 1
#pragma unroll 4
    for (int t = 0; t < N_IMG / 4; ++t) {
        float ya = (float)(4 * t + 2 * khalf) - c;
        float yb = ya + 1.0f;
        ps0 += bilin(img, fmaf(co, ya, sy0), fmaf(si, ya, sx0));
        ps1 += bilin(img, fmaf(co, yb, sy0), fmaf(si, yb, sx0));
    }

    // Wave reduction via matrix unit: D = ones(16x4) x B(4x16) + 0.
    // B holds the 4 per-detector partials (2 per lane across the lane halves);
    // every element of D column n becomes the full sum for detector n.
    v2f ones; ones[0] = 1.0f; ones[1] = 1.0f;
    v2f bm;   bm[0]   = ps0;  bm[1]   = ps1;
    v8f acc = {};
    acc = __builtin_amdgcn_wmma_f32_16x16x4_f32(
        /*neg_a=*/false, ones, /*neg_b=*/false, bm,
        /*c_mod=*/(short)0, acc, /*reuse_a=*/false, /*reuse_b=*/false);

    if (lane < 16) {
        out[det * N_ANG + ang] = acc[0];
    }
}

extern "C" void kernel_launch(void* const* d_in, const int* in_sizes, int n_in,
                              void* d_out, int out_size, void* d_ws, size_t ws_size,
                              hipStream_t stream) {
    const float* img    = (const float*)d_in[0];
    const int*   angles = (const int*)d_in[1];
    float*       out    = (float*)d_out;

    const int blocksPerAngle = N_IMG / 128;           // 4
    dim3 grid(N_ANG * blocksPerAngle);                // 720
    dim3 block(256);
    radon_wmma_kernel<<<grid, block, 0, stream>>>(img, angles, out);
}